// InvariantPointAttention_74569222193540
// MI455X (gfx1250) — compile-verified
//
#include <hip/hip_runtime.h>
#include <hip/hip_bf16.h>
#include <math.h>

// ---------------------------------------------------------------------------
// Invariant Point Attention on MI455X (gfx1250), fp32 WMMA 16x16x4 everywhere.
// Bandwidth-bound on edge_features (302 MB, streamed twice with NT hints so
// the 192 MB L2 stays dedicated to the 27 MB attn buffer + small tensors).
// ---------------------------------------------------------------------------

typedef __attribute__((ext_vector_type(2))) float v2f;
typedef __attribute__((ext_vector_type(8))) float v8f;

#define NN   768
#define DD   384
#define EE   128
#define HH   12
#define HDIM 16
#define QPP  4
#define VPP  8
#define CW   2112   // combined width = H*E + H*HD + H*VP*4 = 1536+192+384

// workspace offsets (floats)
#define OFF_S      ((size_t)0)                      // 768*576
#define OFF_VQK    (OFF_S    + (size_t)NN*576)      // 768*288
#define OFF_VV     (OFF_VQK  + (size_t)NN*288)      // 768*288
#define OFF_GK     (OFF_VV   + (size_t)NN*288)      // 12*768*4*3
#define OFF_GV     (OFF_GK   + (size_t)HH*NN*QPP*3) // 12*768*8*3
#define OFF_AGV    (OFF_GV   + (size_t)HH*NN*VPP*3) // 12*768*8*3
#define OFF_COMB   (OFF_AGV  + (size_t)HH*NN*VPP*3) // 768*2112
#define OFF_LOGITS (OFF_COMB + (size_t)NN*CW)       // 12*768*768

__device__ __forceinline__ v8f wmma_f32(v2f a, v2f b, v8f c) {
    return __builtin_amdgcn_wmma_f32_16x16x4_f32(false, a, false, b, (short)0, c,
                                                 false, false);
}

__device__ __forceinline__ float ntload(const float* p) {
    return __builtin_nontemporal_load(p);   // global_load ... th:TH_LOAD_NT
}

// Generic fp32 WMMA GEMM: C[MxN] = A[MxK] @ B[KxN], row-major, M%16==N%16==0, K%4==0.
__global__ void wmma_gemm_f32(const float* __restrict__ A, const float* __restrict__ B,
                              float* __restrict__ C, int M, int N, int K) {
    const int lane = threadIdx.x & 31;
    const int wave = (blockIdx.x * blockDim.x + threadIdx.x) >> 5;
    const int tilesN = N >> 4;
    const int tiles  = (M >> 4) * tilesN;
    if (wave >= tiles) return;                 // wave-uniform
    const int m0 = (wave / tilesN) << 4;
    const int n0 = (wave % tilesN) << 4;
    const int half = lane >> 4, l = lane & 15;
    v8f acc = {};
    for (int k = 0; k < K; k += 4) {
        v2f a, b;
        const float* ap = A + (size_t)(m0 + l) * K + k + 2 * half;
        a.x = ap[0]; a.y = ap[1];
        const float* bp = B + (size_t)(k + 2 * half) * N + n0 + l;
        b.x = bp[0]; b.y = bp[N];
        acc = wmma_f32(a, b, acc);
    }
    float* cp = C + (size_t)(m0 + half * 8) * N + n0 + l;
#pragma unroll
    for (int r = 0; r < 8; ++r) cp[(size_t)r * N] = acc[r];
}

// Rigid-transform k-points (QP=4) and v-points (VP=8): g = R*v + t.
__global__ void transform_kernel(const float* __restrict__ VQK, const float* __restrict__ VVb,
                                 const float* __restrict__ R, const float* __restrict__ t,
                                 float* __restrict__ GK, float* __restrict__ GV) {
    int idx = blockIdx.x * blockDim.x + threadIdx.x;
    if (idx >= NN * HH * (QPP + VPP)) return;
    int pp = idx % 12; int tmp = idx / 12; int h = tmp % HH; int n = tmp / HH;
    const float* Rn = R + n * 9; const float* tn = t + n * 3;
    const float* src; float* dst;
    if (pp < QPP) { // k-point: second half of VQK
        src = VQK + (size_t)n * 288 + 144 + (h * QPP + pp) * 3;
        dst = GK + (((size_t)h * NN + n) * QPP + pp) * 3;
    } else {
        int p = pp - QPP;
        src = VVb + (size_t)n * 288 + (h * VPP + p) * 3;
        dst = GV + (((size_t)h * NN + n) * VPP + p) * 3;
    }
    float vx = src[0], vy = src[1], vz = src[2];
    dst[0] = Rn[0] * vx + Rn[1] * vy + Rn[2] * vz + tn[0];
    dst[1] = Rn[3] * vx + Rn[4] * vy + Rn[5] * vz + tn[1];
    dst[2] = Rn[6] * vx + Rn[7] * vy + Rn[8] * vz + tn[2];
}

// logits[h][i][j] = 0.25 * q_h[i] . k_h[j]   (hd^-0.5 = 0.25)
__global__ void qk_kernel(const float* __restrict__ S, float* __restrict__ logits) {
    const int lane = threadIdx.x & 31;
    const int wave = (blockIdx.x * blockDim.x + threadIdx.x) >> 5; // 12*48*48
    const int h = wave / 2304, rem = wave % 2304;
    const int i0 = (rem / 48) << 4, j0 = (rem % 48) << 4;
    const int half = lane >> 4, l = lane & 15;
    v8f acc = {};
    const float* Q = S + h * HDIM;
    const float* K = S + 192 + h * HDIM;
    for (int kk = 0; kk < HDIM; kk += 4) {
        v2f a, b;
        const float* qp = Q + (size_t)(i0 + l) * 576 + kk + 2 * half;
        a.x = qp[0]; a.y = qp[1];
        const float* kp = K + (size_t)(j0 + l) * 576 + kk + 2 * half;
        b.x = kp[0]; b.y = kp[1];
        acc = wmma_f32(a, b, acc);
    }
    float* dst = logits + ((size_t)h * NN + i0 + half * 8) * NN + j0 + l;
#pragma unroll
    for (int r = 0; r < 8; ++r) dst[(size_t)r * NN] = 0.25f * acc[r];
}

// logits[h][i][j] += sum_e edge[i][j][e] * W_eb[e][h]   (M = i*768+j, K=128, N=16 pad)
// edge loads are non-temporal: 302 MB stream must not evict the attn working set.
__global__ void edge_bias_kernel(const float* __restrict__ edge,
                                 const float* __restrict__ W_eb,
                                 float* __restrict__ logits) {
    __shared__ float web[EE * 16];
    for (int x = threadIdx.x; x < EE * 16; x += blockDim.x) {
        int e = x >> 4, hh = x & 15;
        web[x] = (hh < HH) ? W_eb[e * HH + hh] : 0.0f;
    }
    __syncthreads();
    const int lane = threadIdx.x & 31;
    const int wave = (blockIdx.x * blockDim.x + threadIdx.x) >> 5; // 36864 tiles
    const int i = wave / 48;
    const int j0 = (wave % 48) << 4;
    const int half = lane >> 4, l = lane & 15;
    const size_t m0 = (size_t)wave * 16;  // flat ij base
    v8f acc = {};
    for (int k = 0; k < EE; k += 4) {
        v2f a, b;
        const float* ap = edge + (m0 + l) * EE + k + 2 * half;
        a.x = ntload(ap);
        a.y = ntload(ap + 1);
        b.x = web[(k + 2 * half) * 16 + l];
        b.y = web[(k + 2 * half + 1) * 16 + l];
        acc = wmma_f32(a, b, acc);
    }
    if (l < HH) {
        float* dst = logits + ((size_t)l * NN + i) * NN + j0 + half * 8;
#pragma unroll
        for (int r = 0; r < 8; ++r) dst[r] += acc[r];
    }
}

// Fused point-distance penalty + softmax per (h, i) row. gk is L2-resident (442 KB).
__global__ void softmax_kernel(const float* __restrict__ GK, const float* __restrict__ pw,
                               float* __restrict__ logits) {
    __shared__ float red[256];
    const int hi = blockIdx.x;
    const int h = hi / NN, i = hi % NN;
    const float* gkh = GK + (size_t)h * NN * 12;
    float gki[12];
#pragma unroll
    for (int c = 0; c < 12; ++c) gki[c] = gkh[(size_t)i * 12 + c];
    const float coef = 0.11785113f * pw[h];   // sqrt(2/QP)/6 * point_weight
    const float rs3  = 0.57735027f;           // 3^-0.5
    float* row = logits + (size_t)hi * NN;
    float vals[3];
    float mx = -3.0e38f;
#pragma unroll
    for (int s = 0; s < 3; ++s) {
        int j = threadIdx.x + s * 256;
        const float* gkj = gkh + (size_t)j * 12;
        float dist = 0.f;
#pragma unroll
        for (int p = 0; p < 4; ++p) {
            float dx = gki[p * 3]     - gkj[p * 3];
            float dy = gki[p * 3 + 1] - gkj[p * 3 + 1];
            float dz = gki[p * 3 + 2] - gkj[p * 3 + 2];
            dist += sqrtf(fmaxf(dx * dx + dy * dy + dz * dz, 1e-12f));
        }
        float v = (row[j] - coef * dist) * rs3;
        vals[s] = v; mx = fmaxf(mx, v);
    }
    red[threadIdx.x] = mx; __syncthreads();
    for (int off = 128; off > 0; off >>= 1) {
        if (threadIdx.x < off) red[threadIdx.x] = fmaxf(red[threadIdx.x], red[threadIdx.x + off]);
        __syncthreads();
    }
    const float bmax = red[0]; __syncthreads();
    float sum = 0.f;
#pragma unroll
    for (int s = 0; s < 3; ++s) { vals[s] = __expf(vals[s] - bmax); sum += vals[s]; }
    red[threadIdx.x] = sum; __syncthreads();
    for (int off = 128; off > 0; off >>= 1) {
        if (threadIdx.x < off) red[threadIdx.x] += red[threadIdx.x + off];
        __syncthreads();
    }
    const float inv = 1.0f / red[0];
#pragma unroll
    for (int s = 0; s < 3; ++s) row[threadIdx.x + s * 256] = vals[s] * inv;
}

// scal_out[i][h][d] = sum_j attn[h][i][j] * v[j][h][d]  -> COMB cols 1536..1727
__global__ void scal_out_kernel(const float* __restrict__ attn, const float* __restrict__ S,
                                float* __restrict__ comb) {
    const int lane = threadIdx.x & 31;
    const int wave = (blockIdx.x * blockDim.x + threadIdx.x) >> 5; // 12*48
    const int h = wave / 48, i0 = (wave % 48) << 4;
    const int half = lane >> 4, l = lane & 15;
    v8f acc = {};
    const float* A = attn + ((size_t)h * NN + i0) * NN;
    const float* V = S + 384 + h * HDIM;
    for (int k = 0; k < NN; k += 4) {
        v2f a, b;
        const float* ap = A + (size_t)l * NN + k + 2 * half;
        a.x = ap[0]; a.y = ap[1];
        const float* bp = V + (size_t)(k + 2 * half) * 576 + l;
        b.x = bp[0]; b.y = bp[576];
        acc = wmma_f32(a, b, acc);
    }
    float* dst = comb + (size_t)(i0 + half * 8) * CW + 1536 + h * HDIM + l;
#pragma unroll
    for (int r = 0; r < 8; ++r) dst[(size_t)r * CW] = acc[r];
}

// agv[h][i][c<24] = sum_j attn[h][i][j] * gv_flat[h][j][c]  (N = 24 padded to 32)
__global__ void agv_kernel(const float* __restrict__ attn, const float* __restrict__ GV,
                           float* __restrict__ AGV) {
    const int lane = threadIdx.x & 31;
    const int wave = (blockIdx.x * blockDim.x + threadIdx.x) >> 5; // 12*48*2
    const int nt = wave & 1, w2 = wave >> 1;
    const int h = w2 / 48, i0 = (w2 % 48) << 4;
    const int half = lane >> 4, l = lane & 15;
    const int n24 = nt * 16 + l;
    const bool bv = (n24 < 24);
    v8f acc = {};
    const float* A = attn + ((size_t)h * NN + i0) * NN;
    const float* G = GV + (size_t)h * NN * 24;
    for (int k = 0; k < NN; k += 4) {
        v2f a, b;
        const float* ap = A + (size_t)l * NN + k + 2 * half;
        a.x = ap[0]; a.y = ap[1];
        b.x = bv ? G[(size_t)(k + 2 * half) * 24 + n24] : 0.f;
        b.y = bv ? G[(size_t)(k + 2 * half + 1) * 24 + n24] : 0.f;
        acc = wmma_f32(a, b, acc);
    }
    if (bv) {
        float* dst = AGV + ((size_t)h * NN + i0 + half * 8) * 24 + n24;
#pragma unroll
        for (int r = 0; r < 8; ++r) dst[(size_t)r * 24] = acc[r];
    }
}

// edge_out[i][h][e] = sum_j attn[h][i][j]*edge[i][j][e]. One block per i:
// the 12 attn rows (+4 zero-pad rows) are staged into LDS with coalesced loads,
// so the WMMA A-fragment reads are ds_loads with no lane masking (EXEC all-1s),
// and edge is streamed once, non-temporally. This is the 2nd/last edge pass.
__global__ void edge_out_kernel(const float* __restrict__ attn, const float* __restrict__ edge,
                                float* __restrict__ comb) {
    __shared__ float asb[16 * NN];            // 48 KB: rows = h (12 real, 4 zero)
    const int i = blockIdx.x;
    for (int x = threadIdx.x; x < 16 * NN; x += blockDim.x) {
        int h = x / NN, j = x - h * NN;
        asb[x] = (h < HH) ? attn[((size_t)h * NN + i) * NN + j] : 0.0f;
    }
    __syncthreads();
    const int lane = threadIdx.x & 31;
    const int e0 = (threadIdx.x >> 5) << 4;   // wave-in-block * 16
    const int half = lane >> 4, l = lane & 15;
    v8f acc = {};
    const float* E = edge + (size_t)i * NN * EE;
    const float* arow = &asb[l * NN];
    for (int k = 0; k < NN; k += 4) {
        v2f a, b;
        a.x = arow[k + 2 * half];
        a.y = arow[k + 2 * half + 1];
        const float* bp = E + (size_t)(k + 2 * half) * EE + e0 + l;
        b.x = ntload(bp);
        b.y = ntload(bp + EE);
        acc = wmma_f32(a, b, acc);
    }
#pragma unroll
    for (int r = 0; r < 8; ++r) {
        int h = half * 8 + r;
        if (h < HH) comb[(size_t)i * CW + h * EE + e0 + l] = acc[r];
    }
}

// loc = R^T (agv - t); write [nrm, loc] into COMB cols 1728..2111
__global__ void vec_kernel(const float* __restrict__ AGV, const float* __restrict__ R,
                           const float* __restrict__ t, float* __restrict__ comb) {
    int idx = blockIdx.x * blockDim.x + threadIdx.x;
    if (idx >= NN * HH * VPP) return;
    int p = idx & 7; int tmp = idx >> 3; int h = tmp % HH; int n = tmp / HH;
    const float* a = AGV + ((size_t)h * NN + n) * 24 + p * 3;
    const float* Rn = R + n * 9; const float* tn = t + n * 3;
    float vx = a[0] - tn[0], vy = a[1] - tn[1], vz = a[2] - tn[2];
    float lx = Rn[0] * vx + Rn[3] * vy + Rn[6] * vz;
    float ly = Rn[1] * vx + Rn[4] * vy + Rn[7] * vz;
    float lz = Rn[2] * vx + Rn[5] * vy + Rn[8] * vz;
    float nr = sqrtf(fmaxf(lx * lx + ly * ly + lz * lz, 1e-12f));
    float* dst = comb + (size_t)n * CW + 1728 + h * 32 + p * 4;
    dst[0] = nr; dst[1] = lx; dst[2] = ly; dst[3] = lz;
}

// out += node + b_final (residual + bias)
__global__ void epilogue_kernel(const float* __restrict__ node, const float* __restrict__ bf,
                                float* __restrict__ out) {
    int idx = blockIdx.x * blockDim.x + threadIdx.x;
    if (idx >= NN * DD) return;
    out[idx] += node[idx] + bf[idx % DD];
}

extern "C" void kernel_launch(void* const* d_in, const int* in_sizes, int n_in,
                              void* d_out, int out_size, void* d_ws, size_t ws_size,
                              hipStream_t stream) {
    (void)in_sizes; (void)n_in; (void)out_size; (void)ws_size;
    const float* node  = (const float*)d_in[0];
    const float* edge  = (const float*)d_in[1];
    const float* R     = (const float*)d_in[2];
    const float* t     = (const float*)d_in[3];
    const float* Wsqkv = (const float*)d_in[4];
    const float* Wvqk  = (const float*)d_in[5];
    const float* Wvv   = (const float*)d_in[6];
    const float* Web   = (const float*)d_in[7];
    const float* Wfin  = (const float*)d_in[8];
    const float* bfin  = (const float*)d_in[9];
    const float* pw    = (const float*)d_in[10];
    float* out = (float*)d_out;
    float* ws  = (float*)d_ws;

    float* S      = ws + OFF_S;
    float* VQK    = ws + OFF_VQK;
    float* VV     = ws + OFF_VV;
    float* GK     = ws + OFF_GK;
    float* GV     = ws + OFF_GV;
    float* AGV    = ws + OFF_AGV;
    float* COMB   = ws + OFF_COMB;
    float* LOGITS = ws + OFF_LOGITS;

    const int WPB = 4;  // waves per block (blockDim = 128)
    auto blocksFor = [](int waves, int wpb) { return (waves + wpb - 1) / wpb; };

    // 1) projections (WMMA)
    wmma_gemm_f32<<<blocksFor(48 * 36, WPB), 128, 0, stream>>>(node, Wsqkv, S, NN, 576, DD);
    wmma_gemm_f32<<<blocksFor(48 * 18, WPB), 128, 0, stream>>>(node, Wvqk, VQK, NN, 288, DD);
    wmma_gemm_f32<<<blocksFor(48 * 18, WPB), 128, 0, stream>>>(node, Wvv, VV, NN, 288, DD);
    // 2) rigid transforms of points
    transform_kernel<<<(NN * HH * 12 + 255) / 256, 256, 0, stream>>>(VQK, VV, R, t, GK, GV);
    // 3) QK^T per head (WMMA)
    qk_kernel<<<blocksFor(HH * 48 * 48, WPB), 128, 0, stream>>>(S, LOGITS);
    // 4) edge bias (WMMA), first NT pass over edge_features
    edge_bias_kernel<<<blocksFor(36864, WPB), 128, 0, stream>>>(edge, Web, LOGITS);
    // 5) point-distance penalty + softmax
    softmax_kernel<<<HH * NN, 256, 0, stream>>>(GK, pw, LOGITS);
    // 6) attention applications (WMMA); edge_out is second NT pass over edge_features
    scal_out_kernel<<<blocksFor(HH * 48, WPB), 128, 0, stream>>>(LOGITS, S, COMB);
    agv_kernel<<<blocksFor(HH * 48 * 2, WPB), 128, 0, stream>>>(LOGITS, GV, AGV);
    edge_out_kernel<<<NN, 256, 0, stream>>>(LOGITS, edge, COMB);
    // 7) inverse-frame transform + norms
    vec_kernel<<<(NN * HH * VPP + 255) / 256, 256, 0, stream>>>(AGV, R, t, COMB);
    // 8) final projection (WMMA) + residual/bias
    wmma_gemm_f32<<<blocksFor(48 * 24, WPB), 128, 0, stream>>>(COMB, Wfin, out, NN, DD, CW);
    epilogue_kernel<<<(NN * DD + 255) / 256, 256, 0, stream>>>(node, bfin, out);
}